// SoftCrossEntropyLoss_58497454571824
// MI455X (gfx1250) — compile-verified
//
#include <hip/hip_runtime.h>
#include <cstdint>

#define BLOCK  256
#define TILE_F 1024   // floats per tile per matrix (BLOCK * 4)
#define PIPE   4      // tiles in flight (4 buffers, 32 KB LDS for x+t)

// ---------------------------------------------------------------------------
// helpers
// ---------------------------------------------------------------------------
__device__ __forceinline__ unsigned ldsOffsetOf(const void* p) {
  // generic pointer to LDS: low 32 bits are the byte offset within the
  // workgroup's LDS allocation (what VDST of the async op needs).
  return (unsigned)(unsigned long long)(uintptr_t)p;
}

// GVS-mode async copy: copies 16 bytes from saddr plus the offset in vaddr to the LDS address in vdst; tracked by ASYNCcnt.
// NT hint: 1 GB stream must not evict the L2-resident weight vector.
__device__ __forceinline__ void asyncCopy16(unsigned lds, unsigned goff, const float* base) {
  asm volatile("global_load_async_to_lds_b128 %0, %1, %2 th:TH_LOAD_NT"
               :: "v"(lds), "v"(goff), "s"(base) : "memory");
}
__device__ __forceinline__ void waitAsync0() { asm volatile("s_wait_asynccnt 0x0" ::: "memory"); }
__device__ __forceinline__ void waitAsync2() { asm volatile("s_wait_asynccnt 0x2" ::: "memory"); }
__device__ __forceinline__ void waitAsync4() { asm volatile("s_wait_asynccnt 0x4" ::: "memory"); }
__device__ __forceinline__ void waitAsync6() { asm volatile("s_wait_asynccnt 0x6" ::: "memory"); }

// fused update: A += t*w*x, B += t*w, online (m, s) for sum-exp
__device__ __forceinline__ void process4(const float4 xv, const float4 tv, const float4 wv,
                                         float& A, float& B, float& m, float& s) {
  const float tw0 = tv.x * wv.x, tw1 = tv.y * wv.y;
  const float tw2 = tv.z * wv.z, tw3 = tv.w * wv.w;
  A = fmaf(tw0, xv.x, A); A = fmaf(tw1, xv.y, A);
  A = fmaf(tw2, xv.z, A); A = fmaf(tw3, xv.w, A);
  B += (tw0 + tw1) + (tw2 + tw3);
  const float m4 = fmaxf(fmaxf(xv.x, xv.y), fmaxf(xv.z, xv.w));
  const float nm = fmaxf(m, m4);
  const float e  = __expf(xv.x - nm) + __expf(xv.y - nm) +
                   __expf(xv.z - nm) + __expf(xv.w - nm);
  s = fmaf(s, __expf(m - nm), e);   // one rescale per 4 elements
  m = nm;
}

__global__ void sce_zero(float* out) { out[0] = 0.0f; }

// ---------------------------------------------------------------------------
// one block per row; 4-deep async pipeline streaming (input, target) via LDS
// ---------------------------------------------------------------------------
__global__ __launch_bounds__(BLOCK)
void sce_kernel(const float* __restrict__ x, const float* __restrict__ t,
                const float* __restrict__ w, float* __restrict__ out,
                int K, float invN) {
  __shared__ float4 xbuf[PIPE][BLOCK];
  __shared__ float4 tbuf[PIPE][BLOCK];
  __shared__ float  redA[BLOCK / 32], redB[BLOCK / 32];
  __shared__ float  redM[BLOCK / 32], redS[BLOCK / 32];

  const int      tid     = threadIdx.x;
  const int      row     = blockIdx.x;
  const unsigned rowByte = (unsigned)((unsigned long long)row * (unsigned long long)K * 4ull);

  float A = 0.0f, B = 0.0f, m = -__builtin_inff(), s = 0.0f;

  const int nTiles = K / TILE_F;
  const int tail   = K - nTiles * TILE_F;
  const float4* w4 = (const float4*)w;

  auto issueTile = [&](int it) {
    const int      b    = it & (PIPE - 1);
    const unsigned goff = rowByte + (unsigned)((it * TILE_F + tid * 4) * 4);
    asyncCopy16(ldsOffsetOf(&xbuf[b][tid]), goff, x);
    asyncCopy16(ldsOffsetOf(&tbuf[b][tid]), goff, t);
  };

  auto consume = [&](int it) {
    const int    b  = it & (PIPE - 1);
    const float4 xv = xbuf[b][tid];      // ds_load_b128
    const float4 tv = tbuf[b][tid];
    process4(xv, tv, w4[it * (TILE_F / 4) + tid], A, B, m, s);
  };

  // prologue: fill the pipe
  const int nPro = (nTiles < PIPE) ? nTiles : PIPE;
  for (int i = 0; i < nPro; ++i) issueTile(i);

  // steady state: newest issued tile is it+PIPE-1 -> wait asynccnt <= 6
  int it = 0;
  for (; it < nTiles - (PIPE - 1); ++it) {
    const float4 wv = w4[it * (TILE_F / 4) + tid];  // overlap L2 hit with wait
    (void)wv;
    waitAsync6();
    consume(it);
    if (it + PIPE < nTiles) issueTile(it + PIPE);
  }
  // drain epilogue: remaining tiles after this one = 2, 1, 0
  for (; it < nTiles; ++it) {
    const int after = nTiles - 1 - it;
    if      (after >= 2) waitAsync4();
    else if (after == 1) waitAsync2();
    else                 waitAsync0();
    consume(it);
  }

  // tail (< TILE_F floats) via direct loads
  if (tid * 4 < tail) {
    const unsigned long long base = (unsigned long long)row * (unsigned long long)K
                                  + (unsigned long long)(nTiles * TILE_F + tid * 4);
    const float4 xv = *(const float4*)(x + base);
    const float4 tv = *(const float4*)(t + base);
    const float4 wv = w4[(nTiles * TILE_F) / 4 + tid];
    process4(xv, tv, wv, A, B, m, s);
  }

  // wave32 butterfly reduction
  for (int off = 16; off > 0; off >>= 1) {
    A += __shfl_xor(A, off, 32);
    B += __shfl_xor(B, off, 32);
    const float mo = __shfl_xor(m, off, 32);
    const float so = __shfl_xor(s, off, 32);
    const float nm = fmaxf(m, mo);
    s = s * __expf(m - nm) + so * __expf(mo - nm);
    m = nm;
  }

  if ((tid & 31) == 0) {
    const int wv_ = tid >> 5;
    redA[wv_] = A; redB[wv_] = B; redM[wv_] = m; redS[wv_] = s;
  }
  __syncthreads();

  if (tid == 0) {
    float a = redA[0], bb = redB[0], mm = redM[0], ss = redS[0];
    for (int i = 1; i < BLOCK / 32; ++i) {
      a += redA[i]; bb += redB[i];
      const float mo = redM[i], so = redS[i];
      const float nm = fmaxf(mm, mo);
      ss = ss * __expf(mm - nm) + so * __expf(mo - nm);
      mm = nm;
    }
    const float loss = (mm + __logf(ss)) * bb - a;   // (m+logS)*B - A
    atomicAdd(out, loss * invN);
  }
}

// ---------------------------------------------------------------------------
extern "C" void kernel_launch(void* const* d_in, const int* in_sizes, int n_in,
                              void* d_out, int out_size, void* d_ws, size_t ws_size,
                              hipStream_t stream) {
  const float* x = (const float*)d_in[0];   // input  [N, K]
  const float* t = (const float*)d_in[1];   // target [N, K]
  const float* w = (const float*)d_in[2];   // weight [K]
  float* out = (float*)d_out;

  const int K = in_sizes[2];
  const int N = in_sizes[0] / K;

  sce_zero<<<1, 1, 0, stream>>>(out);
  sce_kernel<<<N, BLOCK, 0, stream>>>(x, t, w, out, K, 1.0f / (float)N);
}